// HCLGR_19146964205957
// MI455X (gfx1250) — compile-verified
//
#include <hip/hip_runtime.h>

typedef float v2f __attribute__((ext_vector_type(2)));
typedef float v8f __attribute__((ext_vector_type(8)));

#define DIM 64
#define LAYERS 3

// ---------------- vectorized elementwise helpers (float4 / b128 path) ----------------

__global__ void k_zero4(float4* __restrict__ p, int n4) {
    int i = blockIdx.x * blockDim.x + threadIdx.x;
    if (i < n4) p[i] = make_float4(0.f, 0.f, 0.f, 0.f);
}

__global__ void k_copy4(float4* __restrict__ dst, const float4* __restrict__ src, int n4) {
    int i = blockIdx.x * blockDim.x + threadIdx.x;
    if (i < n4) dst[i] = src[i];
}

__global__ void k_add4(float4* __restrict__ dst, const float4* __restrict__ src, int n4) {
    int i = blockIdx.x * blockDim.x + threadIdx.x;
    if (i < n4) {
        float4 a = dst[i], b = src[i];
        a.x += b.x; a.y += b.y; a.z += b.z; a.w += b.w;
        dst[i] = a;
    }
}

// ---------------- edge scatter (segment_sum) ----------------
// dst_emb[seg_idx[e]][d] += vals[e] * src_emb[src_idx[e]][d]
// 16 threads per edge, float4 (b128) gather -> one coalesced 256B row per
// edge, 4 HW float atomics per thread. Tables are L2-resident (192MB L2),
// so minimizing per-edge instruction count is what matters.

__global__ void k_scatter(const float* __restrict__ vals,
                          const int*   __restrict__ seg_idx,
                          const int*   __restrict__ src_idx,
                          const float* __restrict__ src_emb,
                          float*       __restrict__ dst_emb,
                          int nnz) {
    unsigned t = blockIdx.x * blockDim.x + threadIdx.x;   // nnz*16 < 2^31
    unsigned e = t >> 4;
    unsigned q = (t & 15u) << 2;                          // dim offset: 0,4,...,60
    if (e >= (unsigned)nnz) return;
    float v = vals[e];
    int dseg = seg_idx[e];
    int ssrc = src_idx[e];
    const float4 g = *(const float4*)(src_emb + (size_t)ssrc * DIM + q);
    float* p = dst_emb + (size_t)dseg * DIM + q;
    unsafeAtomicAdd(p + 0, v * g.x);
    unsafeAtomicAdd(p + 1, v * g.y);
    unsafeAtomicAdd(p + 2, v * g.z);
    unsafeAtomicAdd(p + 3, v * g.w);
}

// ---------------- fused GEMM with WMMA f32 16x16x4 ----------------
// msg[m,:] = concat(node_msg[m,:], node_msg[m,:]*ue[m,:]) @ W + bias
// One wave32 computes a 16-row x 64-col tile: 4 accumulators (16x16 f32).
// A layout (16x4 f32): lanes 0-15 hold M=lane, K=k0/k0+1; lanes 16-31 hold
// M=lane-16, K=k0+2/k0+3. B (4x16) mirrors this with N across lanes.
// D layout: VGPR r -> M = r + 8*(lane>=16), N = lane&15.
// In-place safe (ue may alias msg): each wave reads only its own 16 rows
// during the K loop, then stores the same 16 rows afterwards.

__global__ void __launch_bounds__(256) k_gemm(const float* __restrict__ node_msg,
                                              const float* __restrict__ ue,
                                              const float* __restrict__ W,     // 128 x 64 row-major
                                              const float* __restrict__ bias,  // 64
                                              float*       __restrict__ msg,
                                              int m_tiles) {
    const int lane = threadIdx.x & 31;
    const int wave = threadIdx.x >> 5;
    const int tile = blockIdx.x * 8 + wave;
    if (tile >= m_tiles) return;   // wave-uniform exit: EXEC stays all-ones for WMMA

    const int m0   = tile * 16;
    const int half = lane >> 4;    // 0 -> K pair (k0,k0+1); 1 -> (k0+2,k0+3)
    const int l15  = lane & 15;

    const float* nm_row = node_msg + (size_t)(m0 + l15) * DIM;
    const float* ue_row = ue       + (size_t)(m0 + l15) * DIM;

    v8f acc0 = {}, acc1 = {}, acc2 = {}, acc3 = {};

    // First half: K = 0..63 comes straight from node_msg
    #pragma unroll 4
    for (int k0 = 0; k0 < DIM; k0 += 4) {
        const int ka = k0 + half * 2;
        v2f a;
        a.x = nm_row[ka];
        a.y = nm_row[ka + 1];
        const float* w0 = W + (size_t)ka * DIM;
        const float* w1 = w0 + DIM;
        v2f b0, b1, b2, b3;
        b0.x = w0[l15 +  0]; b0.y = w1[l15 +  0];
        b1.x = w0[l15 + 16]; b1.y = w1[l15 + 16];
        b2.x = w0[l15 + 32]; b2.y = w1[l15 + 32];
        b3.x = w0[l15 + 48]; b3.y = w1[l15 + 48];
        acc0 = __builtin_amdgcn_wmma_f32_16x16x4_f32(false, a, false, b0, (short)0, acc0, false, false);
        acc1 = __builtin_amdgcn_wmma_f32_16x16x4_f32(false, a, false, b1, (short)0, acc1, false, false);
        acc2 = __builtin_amdgcn_wmma_f32_16x16x4_f32(false, a, false, b2, (short)0, acc2, false, false);
        acc3 = __builtin_amdgcn_wmma_f32_16x16x4_f32(false, a, false, b3, (short)0, acc3, false, false);
    }

    // Second half: K = 64..127 is node_msg * ue (edge_node_element)
    #pragma unroll 4
    for (int k0 = 0; k0 < DIM; k0 += 4) {
        const int ka = k0 + half * 2;
        v2f a;
        a.x = nm_row[ka]     * ue_row[ka];
        a.y = nm_row[ka + 1] * ue_row[ka + 1];
        const float* w0 = W + (size_t)(DIM + ka) * DIM;
        const float* w1 = w0 + DIM;
        v2f b0, b1, b2, b3;
        b0.x = w0[l15 +  0]; b0.y = w1[l15 +  0];
        b1.x = w0[l15 + 16]; b1.y = w1[l15 + 16];
        b2.x = w0[l15 + 32]; b2.y = w1[l15 + 32];
        b3.x = w0[l15 + 48]; b3.y = w1[l15 + 48];
        acc0 = __builtin_amdgcn_wmma_f32_16x16x4_f32(false, a, false, b0, (short)0, acc0, false, false);
        acc1 = __builtin_amdgcn_wmma_f32_16x16x4_f32(false, a, false, b1, (short)0, acc1, false, false);
        acc2 = __builtin_amdgcn_wmma_f32_16x16x4_f32(false, a, false, b2, (short)0, acc2, false, false);
        acc3 = __builtin_amdgcn_wmma_f32_16x16x4_f32(false, a, false, b3, (short)0, acc3, false, false);
    }

    const float bs0 = bias[l15], bs1 = bias[l15 + 16], bs2 = bias[l15 + 32], bs3 = bias[l15 + 48];
    const int row_base = m0 + half * 8;
    #pragma unroll
    for (int r = 0; r < 8; ++r) {
        float* orow = msg + (size_t)(row_base + r) * DIM + l15;
        orow[0]  = acc0[r] + bs0;
        orow[16] = acc1[r] + bs1;
        orow[32] = acc2[r] + bs2;
        orow[48] = acc3[r] + bs3;
    }
}

// ---------------- launch ----------------

extern "C" void kernel_launch(void* const* d_in, const int* in_sizes, int n_in,
                              void* d_out, int out_size, void* d_ws, size_t ws_size,
                              hipStream_t stream) {
    const float* user_emb = (const float*)d_in[0];
    const float* item_emb = (const float*)d_in[1];
    const float* vals     = (const float*)d_in[2];
    const float* W        = (const float*)d_in[3];
    const float* bias     = (const float*)d_in[4];
    const int*   rows     = (const int*)d_in[5];
    const int*   cols     = (const int*)d_in[6];

    const int n_users = in_sizes[0] / DIM;
    const int n_items = in_sizes[1] / DIM;
    const int nnz     = in_sizes[2];

    float* out_node = (float*)d_out;                        // n_items x DIM (final_node)
    float* out_edge = out_node + (size_t)n_items * DIM;     // n_users x DIM (final_edge)

    float* node_msg = (float*)d_ws;                         // n_users x DIM
    float* msg      = node_msg + (size_t)n_users * DIM;     // n_users x DIM
    float* itbuf    = msg + (size_t)n_users * DIM;          // n_items x DIM

    const int n_user4 = (n_users * DIM) / 4;
    const int n_item4 = (n_items * DIM) / 4;
    const int EW = 256;
    const int gb_user4 = (n_user4 + EW - 1) / EW;
    const int gb_item4 = (n_item4 + EW - 1) / EW;

    // init running sums: final_node = item_emb; final_edge = user_emb
    k_copy4<<<gb_item4, EW, 0, stream>>>((float4*)out_node, (const float4*)item_emb, n_item4);
    k_copy4<<<gb_user4, EW, 0, stream>>>((float4*)out_edge, (const float4*)user_emb, n_user4);

    const unsigned sc_threads = (unsigned)nnz * 16u;
    const unsigned sc_blocks  = (sc_threads + EW - 1) / EW;

    const int m_tiles = (n_users + 15) / 16;
    const int gemm_blocks = (m_tiles + 7) / 8;

    for (int i = 0; i < LAYERS; ++i) {
        const float* ie = (i == 0) ? item_emb : itbuf;
        const float* ue = (i == 0) ? user_emb : msg;

        // node_msg = segment_sum(v * ie[cols], rows)
        k_zero4<<<gb_user4, EW, 0, stream>>>((float4*)node_msg, n_user4);
        k_scatter<<<sc_blocks, EW, 0, stream>>>(vals, rows, cols, ie, node_msg, nnz);

        // msg = concat(node_msg, node_msg*ue) @ W[i] + b[i]   (WMMA f32)
        k_gemm<<<gemm_blocks, EW, 0, stream>>>(node_msg, ue,
                                               W + (size_t)i * 2 * DIM * DIM,
                                               bias + (size_t)i * DIM,
                                               msg, m_tiles);

        // final_edge += msg
        k_add4<<<gb_user4, EW, 0, stream>>>((float4*)out_edge, (const float4*)msg, n_user4);

        // itbuf = segment_sum(v * msg[rows], cols)
        k_zero4<<<gb_item4, EW, 0, stream>>>((float4*)itbuf, n_item4);
        k_scatter<<<sc_blocks, EW, 0, stream>>>(vals, cols, rows, msg, itbuf, nnz);

        // final_node += itbuf
        k_add4<<<gb_item4, EW, 0, stream>>>((float4*)out_node, (const float4*)itbuf, n_item4);
    }
}